// PairProductSpin_56556129354391
// MI455X (gfx1250) — compile-verified
//
#include <hip/hip_runtime.h>

// ---------------------------------------------------------------------------
// Batched 256x256 fp32 determinants via LDS-resident blocked LU (NB=16),
// Schur updates on V_WMMA_F32_16X16X4_F32 with K-blocking: each 16x16 C tile
// is loaded once, accumulated through 4 K=4 WMMAs, stored once. All A/B
// chunks are hoisted into registers before the WMMA chain so one dscnt wait
// covers the whole tile. L21 is stored pre-negated so D = A*B + C implements
// C - L*U with zero per-tile VALU. One workgroup (16 wave32) per determinant;
// matrix lives in the WGP's 320KB LDS (256 x 257-float pitch = 263KB).
// Trailing dims are always multiples of 16 -> branch-free full tiles.
// Pivoting path minimizes __syncthreads: redundant per-thread scan of the
// wave-level argmax results, and a barrier-free per-column triangular solve.
// ---------------------------------------------------------------------------

typedef __attribute__((ext_vector_type(2))) float v2f;
typedef __attribute__((ext_vector_type(8))) float v8f;

#define NFULL    512
#define NSUB     256
#define LDP      257          // pitch in floats, odd (bank-conflict spread)
#define NB       16           // panel width == 4 WMMA K-steps
#define NTHREADS 512
#define NWAVES   (NTHREADS / 32)

__global__ __launch_bounds__(NTHREADS)
void det_lu_wmma_kernel(const float* __restrict__ x,
                        const float* __restrict__ F,
                        float* __restrict__ out) {
  extern __shared__ char smem_raw[];
  float* S    = (float*)smem_raw;                  // NSUB * LDP fp32
  int*   up   = (int*)(S + NSUB * LDP);            // 256 row indices
  int*   dn   = up + NSUB;                         // 256 col indices
  float* redv = (float*)(dn + NSUB);               // NWAVES pivot |values|
  float* reds = redv + NWAVES;                     // NWAVES pivot signed values
  int*   redi = (int*)(reds + NWAVES);             // NWAVES pivot indices

  __shared__ float s_det;
  __shared__ int   s_sign;

  const int b    = blockIdx.x;
  const int tid  = threadIdx.x;
  const int lane = tid & 31;
  const int wave = tid >> 5;

  // ---- build up[] / dn[] index lists (flatnonzero order == ascending) ----
  if (tid == 0) {
    s_det = 1.0f;
    s_sign = 1;
    int nu = 0, nd = 0;
    const float* xb = x + (size_t)b * NFULL;
    for (int i = 0; i < NFULL; ++i) {
      if (xb[i] > 0.0f) up[nu++] = i;
      else              dn[nd++] = i;
    }
  }
  __syncthreads();

  // ---- gather submatrix into LDS: S[i][j] = F_full[up[i]][dn[j]] ----
  // F stores row-major off-diagonals: F_full[r][c] = F[r*511 + (c<r ? c : c-1)]
  for (int e = tid; e < NSUB * NSUB; e += NTHREADS) {
    const int i = e >> 8;
    const int j = e & (NSUB - 1);
    const int r = up[i];
    const int c = dn[j];
    S[i * LDP + j] = (c == r) ? 0.0f : F[(size_t)r * (NFULL - 1) + (c < r ? c : c - 1)];
  }
  __syncthreads();

  // ---- blocked right-looking LU with partial pivoting ----
  for (int k0 = 0; k0 < NSUB; k0 += NB) {
    const int i0 = k0 + NB;

    // ===== panel factorization: columns k0 .. k0+15 =====
    for (int jj = 0; jj < NB; ++jj) {
      const int j = k0 + jj;

      // pivot argmax over rows j..255 of column j (carry signed value too)
      float best = -1.0f, bval = 1.0f;
      int   bidx = j;
      for (int i = j + tid; i < NSUB; i += NTHREADS) {
        const float sv = S[i * LDP + j];
        const float av = fabsf(sv);
        if (av > best) { best = av; bval = sv; bidx = i; }
      }
#pragma unroll
      for (int off = 16; off > 0; off >>= 1) {
        const float ov  = __shfl_xor(best, off, 32);
        const float osv = __shfl_xor(bval, off, 32);
        const int   oi  = __shfl_xor(bidx, off, 32);
        if (ov > best) { best = ov; bval = osv; bidx = oi; }
      }
      if (lane == 0) { redv[wave] = best; reds[wave] = bval; redi[wave] = bidx; }
      __syncthreads();                              // barrier 1

      // every thread redundantly resolves the global argmax (no 2nd barrier)
      float bb = redv[0], pvv = reds[0];
      int   p  = redi[0];
#pragma unroll
      for (int w = 1; w < NWAVES; ++w)
        if (redv[w] > bb) { bb = redv[w]; pvv = reds[w]; p = redi[w]; }
      const float inv = 1.0f / pvv;

      if (tid == 0) {
        s_det *= pvv;                               // det = prod(pivots)
        if (p != j) s_sign = -s_sign;
      }

      if (p != j) {                                 // full row swap
        for (int c = tid; c < NSUB; c += NTHREADS) {
          const float t0 = S[j * LDP + c];
          S[j * LDP + c] = S[p * LDP + c];
          S[p * LDP + c] = t0;
        }
        __syncthreads();                            // barrier 2 (only if swap)
      }

      // scale L column (trailing rows stored NEGATED for the WMMA Schur
      // update); rank-1 update confined to remaining panel columns
      for (int i = j + 1 + tid; i < NSUB; i += NTHREADS) {
        const float l = S[i * LDP + j] * inv;
        S[i * LDP + j] = (i >= i0) ? -l : l;
        for (int c = j + 1; c < k0 + NB; ++c)
          S[i * LDP + c] -= l * S[j * LDP + c];
      }
      __syncthreads();                              // barrier 3
    }

    if (i0 >= NSUB) break;

    // ===== U12 = L11^{-1} * A12: per-column forward substitution =====
    // Each thread owns one column; u[] lives in registers; L11 reads are
    // wave-uniform LDS broadcasts. No barriers inside the solve.
    for (int c = i0 + tid; c < NSUB; c += NTHREADS) {
      float u[NB];
#pragma unroll
      for (int jj2 = 0; jj2 < NB; ++jj2) {
        float acc = S[(k0 + jj2) * LDP + c];
#pragma unroll
        for (int m2 = 0; m2 < jj2; ++m2)
          acc -= S[(k0 + jj2) * LDP + (k0 + m2)] * u[m2];
        u[jj2] = acc;
        S[(k0 + jj2) * LDP + c] = acc;
      }
    }
    __syncthreads();

    // ===== Schur update A22 -= L21 * U12, K-blocked on WMMA f32 16x16x4 =====
    // L21 pre-negated -> plain loads feed D = A*B + C directly. All operand
    // loads are hoisted ahead of the back-to-back WMMA chain.
    const int T     = NSUB - i0;       // always a multiple of 16
    const int nt    = T >> 4;
    const int total = nt * nt;
    const int m  = lane & 15;
    const int kh = lane >> 4;          // half-wave selector

    for (int t = wave; t < total; t += NWAVES) {    // wave-uniform loop
      const int tr = t / nt, tc = t % nt;
      const int rowBase = i0 + tr * 16;
      const int colBase = i0 + tc * 16;

      // C 16x16: VGPR r -> row rowBase + r + 8*kh, col colBase + m
      float* Cp = S + (rowBase + 8 * kh) * LDP + colBase + m;
      v8f c;
#pragma unroll
      for (int r = 0; r < 8; ++r)
        c[r] = Cp[r * LDP];

      // hoist all 4 A-chunks and 4 B-chunks into registers
      v2f a[NB / 4], bv[NB / 4];
      const float* Ar = S + (rowBase + m) * LDP + k0 + 2 * kh;  // + 4*kk
      const float* Bc = S + (k0 + kh) * LDP + colBase + m;      // + 4*kk*LDP
#pragma unroll
      for (int kk = 0; kk < NB / 4; ++kk) {
        // A (pre-negated L21) 16x4: lane m -> row rowBase+m; VGPR v -> K=v+2*kh
        a[kk].x = Ar[4 * kk];
        a[kk].y = Ar[4 * kk + 1];
        // B (U12) 4x16: lane m -> col colBase+m; VGPR v -> K = 2*v+kh
        bv[kk].x = Bc[4 * kk * LDP];
        bv[kk].y = Bc[(4 * kk + 2) * LDP];
      }

#pragma unroll
      for (int kk = 0; kk < NB / 4; ++kk)
        c = __builtin_amdgcn_wmma_f32_16x16x4_f32(
                /*neg_a=*/false, a[kk], /*neg_b=*/false, bv[kk],
                /*c_mod=*/(short)0, c, /*reuse_a=*/false, /*reuse_b=*/false);

#pragma unroll
      for (int r = 0; r < 8; ++r)
        Cp[r * LDP] = c[r];
    }
    __syncthreads();
  }

  if (tid == 0) out[b] = s_det * (float)s_sign;
}

extern "C" void kernel_launch(void* const* d_in, const int* in_sizes, int n_in,
                              void* d_out, int out_size, void* d_ws, size_t ws_size,
                              hipStream_t stream) {
  const float* x = (const float*)d_in[0];   // (B, 512) fp32
  const float* F = (const float*)d_in[1];   // (512*511,) fp32 off-diagonals
  float* out = (float*)d_out;               // (B,) fp32 determinants

  const int B = in_sizes[0] / NFULL;        // = 1024

  const size_t shmem = (size_t)(NSUB * LDP) * sizeof(float)   // matrix (~263 KB)
                     + (size_t)(2 * NSUB) * sizeof(int)       // up/dn lists
                     + (size_t)(2 * NWAVES) * sizeof(float)   // pivot red vals
                     + (size_t)NWAVES * sizeof(int);          // pivot red idx

  det_lu_wmma_kernel<<<B, NTHREADS, shmem, stream>>>(x, F, out);
}